// S6_62156766707948
// MI455X (gfx1250) — compile-verified
//
#include <hip/hip_runtime.h>
#include <hip/hip_bf16.h>
#include <math.h>

#define L_SEQ 4096
#define C_CH  1024
#define S_ST  16
#define NCH   64            // chunks along L
#define CHUNK (L_SEQ/NCH)   // 64 sequential steps per chunk

typedef float v2f __attribute__((ext_vector_type(2)));
typedef float v8f __attribute__((ext_vector_type(8)));

// dt_bias = log(exp(0.01)-1)
#define DT_BIAS (-4.60016602f)

// ---------------------------------------------------------------------------
// Kernel 1: B = x@W_B + b_B, Cm = x@W_C + b_C  via V_WMMA_F32_16X16X4_F32.
// One wave per 16-row L-tile; N = S = 16 exactly one WMMA tile wide.
// A fragment (16x4 f32): lanes 0-15 hold {K=0,K=1}, lanes 16-31 hold {K=2,K=3}.
// B fragment (4x16 f32): mirrored — lane n holds {K rows} for column n.
// D (16x16 f32, 8 VGPRs): VGPR r -> M=r (lanes 0-15) / M=8+r (lanes 16-31).
// ---------------------------------------------------------------------------
__global__ __launch_bounds__(32)
void s6_proj_wmma(const float* __restrict__ x,
                  const float* __restrict__ W_B, const float* __restrict__ b_B,
                  const float* __restrict__ W_C, const float* __restrict__ b_C,
                  float* __restrict__ Bc, float* __restrict__ Cc) {
  const int tile = blockIdx.x;          // 256 tiles of 16 rows of L
  const int lane = threadIdx.x;         // 0..31
  const int half = lane >> 4;           // 0: K=0,1 | 1: K=2,3
  const int mn   = lane & 15;           // A row / B column
  const int l0   = tile * 16;

  v8f accB = {};
  v8f accC = {};
  for (int k = 0; k < C_CH; k += 4) {
    const int ka = k + (half << 1);
    const float* xp = x + (size_t)(l0 + mn) * C_CH + ka;
    v2f a;  a.x  = xp[0];                          a.y  = xp[1];
    v2f bb; bb.x = W_B[(size_t)ka * S_ST + mn];    bb.y = W_B[(size_t)(ka + 1) * S_ST + mn];
    v2f bc; bc.x = W_C[(size_t)ka * S_ST + mn];    bc.y = W_C[(size_t)(ka + 1) * S_ST + mn];
    accB = __builtin_amdgcn_wmma_f32_16x16x4_f32(false, a, false, bb, (short)0, accB, false, false);
    accC = __builtin_amdgcn_wmma_f32_16x16x4_f32(false, a, false, bc, (short)0, accC, false, false);
  }
  const float biasB = b_B[mn];
  const float biasC = b_C[mn];
#pragma unroll
  for (int r = 0; r < 8; ++r) {
    const int m = half * 8 + r;
    Bc[(size_t)(l0 + m) * S_ST + mn] = accB[r] + biasB;
    Cc[(size_t)(l0 + m) * S_ST + mn] = accC[r] + biasC;
  }
}

// ---------------------------------------------------------------------------
// Kernel 2: u[l] = x[l,:] . W_dt1 + b_dt1   (rank-1; dt is reconstructed on
// the fly in the scan kernels, avoiding the 16 MB (L,C) dt intermediate).
// ---------------------------------------------------------------------------
__global__ __launch_bounds__(256)
void s6_dtproj(const float* __restrict__ x, const float* __restrict__ W_dt1,
               const float* __restrict__ b_dt1, float* __restrict__ u) {
  __shared__ float red[8];
  const int l = blockIdx.x;
  const int t = threadIdx.x;
  float s = 0.f;
  for (int c = t; c < C_CH; c += 256) s += x[(size_t)l * C_CH + c] * W_dt1[c];
  for (int m = 16; m > 0; m >>= 1) s += __shfl_xor(s, m, 32);
  if ((t & 31) == 0) red[t >> 5] = s;
  __syncthreads();
  if (t < 8) {
    float v = red[t];
    v += __shfl_xor(v, 4, 32);
    v += __shfl_xor(v, 2, 32);
    v += __shfl_xor(v, 1, 32);
    if (t == 0) u[l] = v + b_dt1[0];
  }
}

__device__ __forceinline__ float s6_softplus(float z) {
  return (z > 20.f) ? z : log1pf(__expf(z));
}

// ---------------------------------------------------------------------------
// Kernel 3 (pass A): per-chunk local scan -> carries (prod At, h_end).
// Thread (c,s) lane: s = tid&15, c = 16*blockIdx.y + tid/16. 64 serial steps.
// ---------------------------------------------------------------------------
__global__ __launch_bounds__(256)
void s6_scan_carry(const float* __restrict__ x, const float* __restrict__ Bc,
                   const float* __restrict__ u, const float* __restrict__ lognegA,
                   const float* __restrict__ W_dt2, const float* __restrict__ b_dt2,
                   float* __restrict__ carryA, float* __restrict__ carryH) {
  const int chunk = blockIdx.x;
  const int t = threadIdx.x;
  const int s = t & 15;
  const int c = blockIdx.y * 16 + (t >> 4);
  const float A    = -__expf(lognegA[(size_t)c * S_ST + s]);
  const float invA = 1.0f / A;
  const float w2   = W_dt2[c];
  const float bb2  = b_dt2[c];
  float h = 0.f, Ap = 1.f;
  const int l0 = chunk * CHUNK;
  for (int i = 0; i < CHUNK; ++i) {
    const int l = l0 + i;
    const float xv = x[(size_t)l * C_CH + c];
    const float Bv = Bc[(size_t)l * S_ST + s];
    const float dt = s6_softplus(DT_BIAS + u[l] * w2 + bb2);
    const float At = __expf(A * dt);
    const float ut = (At - 1.f) * Bv * invA * xv;
    h  = fmaf(At, h, ut);
    Ap *= At;
  }
  const size_t idx = (size_t)chunk * (C_CH * S_ST) + (size_t)c * S_ST + s;
  carryA[idx] = Ap;
  carryH[idx] = h;
}

// ---------------------------------------------------------------------------
// Kernel 4: serial combine across the 64 chunk carries for each of the
// 16384 independent (c,s) scan lanes; emits h_init per chunk.
// ---------------------------------------------------------------------------
__global__ __launch_bounds__(256)
void s6_combine(const float* __restrict__ carryA, const float* __restrict__ carryH,
                float* __restrict__ hinit) {
  const int i = blockIdx.x * 256 + threadIdx.x;   // 0..16383, coalesced
  float h = 0.f;
  for (int k = 0; k < NCH; ++k) {
    const size_t idx = (size_t)k * (C_CH * S_ST) + i;
    hinit[idx] = h;
    h = fmaf(carryA[idx], h, carryH[idx]);
  }
}

// ---------------------------------------------------------------------------
// Kernel 5 (pass C): recompute local scan seeded with h_init; reduce over the
// 16 state lanes (shfl_xor stays inside each 16-lane group of the wave32) and
// write y[l,c].
// ---------------------------------------------------------------------------
__global__ __launch_bounds__(256)
void s6_scan_final(const float* __restrict__ x, const float* __restrict__ Bc,
                   const float* __restrict__ Cc, const float* __restrict__ u,
                   const float* __restrict__ lognegA, const float* __restrict__ W_dt2,
                   const float* __restrict__ b_dt2, const float* __restrict__ hinit,
                   float* __restrict__ y) {
  const int chunk = blockIdx.x;
  const int t = threadIdx.x;
  const int s = t & 15;
  const int c = blockIdx.y * 16 + (t >> 4);
  const float A    = -__expf(lognegA[(size_t)c * S_ST + s]);
  const float invA = 1.0f / A;
  const float w2   = W_dt2[c];
  const float bb2  = b_dt2[c];
  float h = hinit[(size_t)chunk * (C_CH * S_ST) + (size_t)c * S_ST + s];
  const int l0 = chunk * CHUNK;
  for (int i = 0; i < CHUNK; ++i) {
    const int l = l0 + i;
    const float xv = x[(size_t)l * C_CH + c];
    const float Bv = Bc[(size_t)l * S_ST + s];
    const float Cv = Cc[(size_t)l * S_ST + s];
    const float dt = s6_softplus(DT_BIAS + u[l] * w2 + bb2);
    const float At = __expf(A * dt);
    const float ut = (At - 1.f) * Bv * invA * xv;
    h = fmaf(At, h, ut);
    float p = Cv * h;
    p += __shfl_xor(p, 1, 32);
    p += __shfl_xor(p, 2, 32);
    p += __shfl_xor(p, 4, 32);
    p += __shfl_xor(p, 8, 32);
    if (s == 0) y[(size_t)l * C_CH + c] = p;
  }
}

// ---------------------------------------------------------------------------
extern "C" void kernel_launch(void* const* d_in, const int* in_sizes, int n_in,
                              void* d_out, int out_size, void* d_ws, size_t ws_size,
                              hipStream_t stream) {
  const float* x       = (const float*)d_in[0];
  const float* lognegA = (const float*)d_in[1];
  const float* W_B     = (const float*)d_in[2];
  const float* b_B     = (const float*)d_in[3];
  const float* W_C     = (const float*)d_in[4];
  const float* b_C     = (const float*)d_in[5];
  const float* W_dt1   = (const float*)d_in[6];
  const float* b_dt1   = (const float*)d_in[7];
  const float* W_dt2   = (const float*)d_in[8];
  const float* b_dt2   = (const float*)d_in[9];
  float* y = (float*)d_out;

  // workspace layout (floats)
  float* ws = (float*)d_ws;
  float* Bc     = ws;                    // L*S          = 65536
  float* Cc     = Bc + (size_t)L_SEQ * S_ST;
  float* u      = Cc + (size_t)L_SEQ * S_ST;     // L  = 4096
  float* carryA = u + L_SEQ;             // NCH*C*S      = 1048576
  float* carryH = carryA + (size_t)NCH * C_CH * S_ST;
  float* hinit  = carryH + (size_t)NCH * C_CH * S_ST;
  (void)in_sizes; (void)n_in; (void)out_size; (void)ws_size;

  // 1) projections (WMMA f32) — 256 L-tiles, one wave each
  s6_proj_wmma<<<L_SEQ / 16, 32, 0, stream>>>(x, W_B, b_B, W_C, b_C, Bc, Cc);
  // 2) rank-1 dt projection
  s6_dtproj<<<L_SEQ, 256, 0, stream>>>(x, W_dt1, b_dt1, u);
  // 3) chunked scan, pass A: carries
  dim3 gscan(NCH, C_CH / 16);
  s6_scan_carry<<<gscan, 256, 0, stream>>>(x, Bc, u, lognegA, W_dt2, b_dt2,
                                           carryA, carryH);
  // 4) combine chunk carries -> per-chunk h_init
  s6_combine<<<(C_CH * S_ST) / 256, 256, 0, stream>>>(carryA, carryH, hinit);
  // 5) pass C: seeded rescan + output reduction
  s6_scan_final<<<gscan, 256, 0, stream>>>(x, Bc, Cc, u, lognegA, W_dt2, b_dt2,
                                           hinit, y);
}